// mlp_GCN_45105746543003
// MI455X (gfx1250) — compile-verified
//
#include <hip/hip_runtime.h>
#include <hip/hip_bf16.h>
#include <math.h>

// ---------------------------------------------------------------------------
// GCN forward on MI455X (gfx1250, wave32).
// GEMMs use V_WMMA_F32_16X16X4_F32 (full fp32 precision -- compute is trivial
// vs the L2-resident edge aggregation, so no reason to quantize).
// ---------------------------------------------------------------------------

typedef __attribute__((ext_vector_type(2))) float v2f;
typedef __attribute__((ext_vector_type(8))) float v8f;

#define N_NODES 50000
#define N_EDGES 800000
#define D_FEAT  128

// ---------------------------------------------------------------------------
// Degree / normalization
// ---------------------------------------------------------------------------
__global__ void gcn_deg_init(float* __restrict__ deg, int n) {
    int i = blockIdx.x * blockDim.x + threadIdx.x;
    if (i < n) deg[i] = 1.0f;                 // self-loop contributes 1
}

__global__ void gcn_deg_edges(const int* __restrict__ col, float* __restrict__ deg, int e) {
    int i = blockIdx.x * blockDim.x + threadIdx.x;
    if (i < e) atomicAdd(&deg[col[i]], 1.0f);
}

__global__ void gcn_dinv(float* __restrict__ deg, int n) {
    int i = blockIdx.x * blockDim.x + threadIdx.x;
    if (i < n) deg[i] = rsqrtf(deg[i]);       // deg >= 1 always
}

// ---------------------------------------------------------------------------
// Tiled fp32 WMMA GEMM:  OUT[M,N] = act(X[M,128] @ W[128,N] + bias)
//   block = 256 threads = 8 wave32s, each block covers 128 rows,
//   wave w owns rows [blk*128 + w*16, +16), full N via N/16 accumulators.
//
// W is staged into LDS K-pair-interleaved:
//   sW[((k>>1)*N + col)*2 + 0] = W[k  ][col]
//   sW[((k>>1)*N + col)*2 + 1] = W[k+1][col]
// so each B fragment (per ISA 32-bit A/B layout: lane-half selects the K-pair)
// is a single aligned ds_load_b64 straight into an even VGPR pair.
//
// A rows are clamped (min(row, M-1)) instead of guarded: out-of-range A rows
// only feed D rows that are never stored, so no EXEC juggling in the k-loop.
//
// ACT: 0 = none, 1 = relu, 2 = sigmoid(tanh(x))
// ---------------------------------------------------------------------------
template <int N, int ACT>
__global__ void gcn_gemm_wmma(const float* __restrict__ X,
                              const float* __restrict__ W,
                              const float* __restrict__ bias,
                              float* __restrict__ OUT, int M) {
    __shared__ float sW[128 * N];              // 64KB for N=128 (WGP has 320KB)

    const int t = threadIdx.x;                 // 0..255
    // cooperative pair-interleaved stage of W (64*N K-pairs)
    for (int i = t; i < 64 * N; i += 256) {
        const int kp  = i / N;                 // K-pair index 0..63
        const int c   = i % N;
        sW[i * 2 + 0] = W[(2 * kp    ) * N + c];
        sW[i * 2 + 1] = W[(2 * kp + 1) * N + c];
    }
    __syncthreads();

    const int wave    = t >> 5;                // 0..7
    const int lane    = t & 31;
    const int half    = lane >> 4;             // 0 or 1 (lane group)
    const int l16     = lane & 15;
    const int rowBase = blockIdx.x * 128 + wave * 16;

    // clamped A-row pointer (always in-bounds; padded rows never stored)
    const int arow = min(rowBase + l16, M - 1);
    const float* __restrict__ aRow = X + (size_t)arow * 128;

    v8f zero = {};
    v8f acc[N / 16];
#pragma unroll
    for (int nt = 0; nt < N / 16; ++nt) acc[nt] = zero;

    for (int k0 = 0; k0 < 128; k0 += 4) {
        // A fragment (16x4 fp32): lanes = M rows, VGPR pair = K, half -> K+2
        const int kb = k0 + half * 2;
        const v2f a  = *(const v2f*)&aRow[kb];           // global_load_b64

        // B pair row: half=0 -> K={k0,k0+1}, half=1 -> K={k0+2,k0+3}
        const int kp = (k0 >> 1) + half;
        const float* __restrict__ bRow = &sW[(size_t)kp * N * 2];
#pragma unroll
        for (int nt = 0; nt < N / 16; ++nt) {
            const v2f b = *(const v2f*)&bRow[(nt * 16 + l16) * 2];  // ds_load_b64
            acc[nt] = __builtin_amdgcn_wmma_f32_16x16x4_f32(
                false, a, false, b, (short)0, acc[nt], false, false);
        }
    }

    // Epilogue: C/D layout -> VGPR v holds row (v + 8*half), lane l16 = col
#pragma unroll
    for (int v = 0; v < 8; ++v) {
        const int row = rowBase + v + half * 8;
        if (row >= M) continue;
#pragma unroll
        for (int nt = 0; nt < N / 16; ++nt) {
            const int col = nt * 16 + l16;
            float val = acc[nt][v];
            if (bias) val += bias[col];
            if (ACT == 1) {
                val = fmaxf(val, 0.0f);
            } else if (ACT == 2) {
                val = tanhf(val);
                val = 1.0f / (1.0f + __expf(-val));
            }
            OUT[(size_t)row * N + col] = val;
        }
    }
}

// ---------------------------------------------------------------------------
// Aggregation: out = bias + self_loop + scatter_add(edges)
// ---------------------------------------------------------------------------
__global__ void gcn_agg_init(const float* __restrict__ h,
                             const float* __restrict__ dinv,
                             const float* __restrict__ bias,
                             float* __restrict__ out, int total) {
    int i = blockIdx.x * blockDim.x + threadIdx.x;
    if (i >= total) return;
    const int node = i >> 7;
    const int f    = i & 127;
    float w = dinv[node];
    w *= w;                                    // self-loop norm = dinv^2
    out[i] = bias[f] + h[i] * w;
}

__global__ void gcn_agg_edges(const float* __restrict__ h,
                              const float* __restrict__ dinv,
                              const int* __restrict__ row,
                              const int* __restrict__ col,
                              float* __restrict__ out, int e) {
    const int tid  = blockIdx.x * blockDim.x + threadIdx.x;
    const int edge = tid >> 5;                 // 32 lanes per edge
    if (edge >= e) return;
    const int lane = tid & 31;
    const int r = row[edge];
    __builtin_prefetch(h + (size_t)r * 128, 0, 3);   // global_prefetch_b8
    const int c = col[edge];
    const float w = dinv[r] * dinv[c];
    const float4 v = ((const float4*)(h + (size_t)r * 128))[lane];
    float* dst = out + (size_t)c * 128 + lane * 4;
    atomicAdd(dst + 0, v.x * w);
    atomicAdd(dst + 1, v.y * w);
    atomicAdd(dst + 2, v.z * w);
    atomicAdd(dst + 3, v.w * w);
}

// ---------------------------------------------------------------------------
// BatchNorm (training-mode math): per-feature mean/var over 50000 nodes.
// Launch exactly 256 blocks x 256 threads -> stride 65536 is a multiple of
// 128, so each thread's feature index (flat & 127) is constant -> private
// register reduction + one pair of global atomics per thread.
// ---------------------------------------------------------------------------
__global__ void gcn_bn_zero(float* __restrict__ sums) {
    if (threadIdx.x < 256) sums[threadIdx.x] = 0.0f;   // [0,128)=sum, [128,256)=sumsq
}

__global__ void gcn_bn_stats(const float* __restrict__ h,
                             float* __restrict__ sums, int total) {
    const int t      = blockIdx.x * blockDim.x + threadIdx.x;
    const int stride = gridDim.x * blockDim.x;          // multiple of 128
    const int f      = t & 127;
    float s = 0.0f, s2 = 0.0f;
    for (int i = t; i < total; i += stride) {
        const float v = h[i];
        s  += v;
        s2 += v * v;
    }
    atomicAdd(&sums[f], s);
    atomicAdd(&sums[f + 128], s2);
}

__global__ void gcn_bn_finalize(const float* __restrict__ sums,
                                const float* __restrict__ gamma,
                                const float* __restrict__ beta,
                                float* __restrict__ scale,
                                float* __restrict__ shift, float invN) {
    const int f = threadIdx.x;                          // 128 threads
    const float mean = sums[f] * invN;
    const float var  = sums[f + 128] * invN - mean * mean;
    const float sc   = gamma[f] * rsqrtf(var + 1e-5f);
    scale[f] = sc;
    shift[f] = beta[f] - mean * sc;
}

__global__ void gcn_bn_apply_relu(float* __restrict__ h,
                                  const float* __restrict__ scale,
                                  const float* __restrict__ shift, int total) {
    int i = blockIdx.x * blockDim.x + threadIdx.x;
    if (i >= total) return;
    const int f = i & 127;
    h[i] = fmaxf(h[i] * scale[f] + shift[f], 0.0f);
}

__global__ void gcn_relu(float* __restrict__ h, int total) {
    int i = blockIdx.x * blockDim.x + threadIdx.x;
    if (i < total) h[i] = fmaxf(h[i], 0.0f);
}

// ---------------------------------------------------------------------------
// Host-side launch
// ---------------------------------------------------------------------------
extern "C" void kernel_launch(void* const* d_in, const int* in_sizes, int n_in,
                              void* d_out, int out_size, void* d_ws, size_t ws_size,
                              hipStream_t stream) {
    const float* x       = (const float*)d_in[0];
    const int*   eidx    = (const int*)d_in[1];
    const float* pre_w1  = (const float*)d_in[2];
    const float* pre_b1  = (const float*)d_in[3];
    const float* pre_w2  = (const float*)d_in[4];
    const float* pre_b2  = (const float*)d_in[5];
    const float* conv_ws = (const float*)d_in[6];   // [3,128,128]
    const float* conv_bs = (const float*)d_in[7];   // [3,128]
    const float* bn_g    = (const float*)d_in[8];   // [2,128]
    const float* bn_b    = (const float*)d_in[9];   // [2,128]
    const float* post_w  = (const float*)d_in[10];  // [128,64]
    const float* post_b  = (const float*)d_in[11];  // [64]
    float*       out     = (float*)d_out;           // [50000,64]

    const int* row = eidx;
    const int* col = eidx + N_EDGES;

    const int n     = N_NODES;
    const int e     = N_EDGES;
    const int total = n * D_FEAT;                   // 6.4M floats

    // workspace carve (floats)
    float* ws    = (float*)d_ws;
    float* buf0  = ws;                              // 6.4M
    float* buf1  = ws +  6400000;                   // 6.4M
    float* buf2  = ws + 12800000;                   // 6.4M
    float* dinv  = ws + 19200000;                   // 50000
    float* sums  = ws + 19250048;                   // 256 (sum | sumsq)
    float* scale = sums + 256;                      // 128
    float* shift = scale + 128;                     // 128
    (void)ws_size; (void)n_in; (void)in_sizes; (void)out_size;

    const int T = 256;

    // 1) degrees -> dinv
    gcn_deg_init <<<(n + T - 1) / T, T, 0, stream>>>(dinv, n);
    gcn_deg_edges<<<(e + T - 1) / T, T, 0, stream>>>(col, dinv, e);
    gcn_dinv     <<<(n + T - 1) / T, T, 0, stream>>>(dinv, n);

    const int gemmBlocks = (n + 127) / 128;         // 391
    const int elemBlocks = (total + T - 1) / T;     // 25000
    const int edgeBlocks = (e * 32) / T;            // 100000

    // 2) pre-MLP (relu(X W1 + b1), relu(· W2 + b2))
    gcn_gemm_wmma<128, 1><<<gemmBlocks, T, 0, stream>>>(x,    pre_w1, pre_b1, buf0, n);
    gcn_gemm_wmma<128, 1><<<gemmBlocks, T, 0, stream>>>(buf0, pre_w2, pre_b2, buf1, n);

    float* h   = buf1;
    float* tmp = buf0;
    float* agg = buf2;

    // 3) three GCNConv layers
    for (int layer = 0; layer < 3; ++layer) {
        const float* W = conv_ws + (size_t)layer * 128 * 128;
        const float* b = conv_bs + (size_t)layer * 128;

        gcn_gemm_wmma<128, 0><<<gemmBlocks, T, 0, stream>>>(h, W, nullptr, tmp, n);
        gcn_agg_init <<<elemBlocks, T, 0, stream>>>(tmp, dinv, b, agg, total);
        gcn_agg_edges<<<edgeBlocks, T, 0, stream>>>(tmp, dinv, row, col, agg, e);

        if (layer < 2) {
            gcn_bn_zero    <<<1, 256, 0, stream>>>(sums);
            gcn_bn_stats   <<<256, 256, 0, stream>>>(agg, sums, total);
            gcn_bn_finalize<<<1, 128, 0, stream>>>(sums, bn_g + layer * 128,
                                                   bn_b + layer * 128,
                                                   scale, shift, 1.0f / (float)n);
            gcn_bn_apply_relu<<<elemBlocks, T, 0, stream>>>(agg, scale, shift, total);
        } else {
            gcn_relu<<<elemBlocks, T, 0, stream>>>(agg, total);
        }
        // rotate: aggregated output becomes next layer's input
        float* t2 = h; h = agg; agg = t2;
    }

    // 4) post-MLP: sigmoid(tanh(h @ post_w + post_b)), N = 64
    gcn_gemm_wmma<64, 2><<<gemmBlocks, T, 0, stream>>>(h, post_w, post_b, out, n);
}